// W2L_10316511445059
// MI455X (gfx1250) — compile-verified
//
#include <hip/hip_runtime.h>

#define NEG (-1e30f)

typedef __attribute__((ext_vector_type(4))) unsigned int v4u;
typedef __attribute__((ext_vector_type(8))) unsigned int v8u;

// ---------------------------------------------------------------------------
// Problem constants (match reference setup_inputs)
// ---------------------------------------------------------------------------
#define B_  64
#define T_  1000
#define V_  64
#define L_  150
#define S_  301          // 2*L+1
#define CHUNK 16         // time rows per TDM tile

// ---------------------------------------------------------------------------
// TDM: issue a 2-D tensor_load_to_lds of `rows` x 64 f32 from gptr into LDS.
// Descriptor per CDNA5 ISA (groups 0/1 only -> <=2D).
// ---------------------------------------------------------------------------
__device__ __forceinline__ void tdm_load_rows(const float* gptr,
                                              unsigned lds_addr, int rows) {
  unsigned long long ga = (unsigned long long)(__SIZE_TYPE__)gptr;
  v4u g0;
  g0[0] = 1u;                                   // count=1 (valid user D#)
  g0[1] = lds_addr;                             // lds_addr[31:0]
  g0[2] = (unsigned)(ga & 0xffffffffu);         // global_addr[31:0]
  g0[3] = ((unsigned)(ga >> 32) & 0x01ffffffu)  // global_addr[56:32]
          | 0x80000000u;                        // type=2 at bits[127:126]
  v8u g1;
  g1[0] = 0x00020000u;                          // wg_mask=0, data_size=4B
  g1[1] = (unsigned)V_ << 16;                   // tensor_dim0[15:0] @ [63:48]
  g1[2] = ((unsigned)rows & 0xffffu) << 16;     // tensor_dim1[15:0] @ [95:80]
  g1[3] = (unsigned)V_ << 16;                   // tile_dim0 @ [127:112]
  g1[4] = (unsigned)rows & 0xffffu;             // tile_dim1 @ [143:128]
  g1[5] = (unsigned)V_;                         // tensor_dim0_stride[31:0]
  g1[6] = 0u;
  g1[7] = 0u;
  asm volatile("tensor_load_to_lds %0, %1" :: "s"(g0), "s"(g1) : "memory");
}

// wave32 XOR-butterfly via ds_swizzle; offset must be an ICE -> template arg.
// group-of-32 mode: offset = {0, xor[14:10], or[9:5], and[4:0]}
template <int IMM>
__device__ __forceinline__ float swz_f32(float x) {
  int i = __builtin_amdgcn_ds_swizzle(__float_as_int(x), IMM);
  return __int_as_float(i);
}

// ---------------------------------------------------------------------------
// Kernel A: per-row log-sum-exp over V=64 logits. One wave32 per row,
// two elements per lane, ds_swizzle butterfly reductions.
// ---------------------------------------------------------------------------
__global__ void W2L_lse_kernel(const float* __restrict__ logits,
                               float* __restrict__ lse) {
  const int wid  = (int)(blockIdx.x * (blockDim.x >> 5) + (threadIdx.x >> 5));
  const int lane = (int)(threadIdx.x & 31);
  if (wid >= B_ * T_) return;
  const float* row = logits + (size_t)wid * V_;
  float x0 = row[lane];
  float x1 = row[lane + 32];
  float m = fmaxf(x0, x1);
  m = fmaxf(m, swz_f32<0x401f>(m));   // xor 16
  m = fmaxf(m, swz_f32<0x201f>(m));   // xor 8
  m = fmaxf(m, swz_f32<0x101f>(m));   // xor 4
  m = fmaxf(m, swz_f32<0x081f>(m));   // xor 2
  m = fmaxf(m, swz_f32<0x041f>(m));   // xor 1
  float s = __expf(x0 - m) + __expf(x1 - m);
  s += swz_f32<0x401f>(s);
  s += swz_f32<0x201f>(s);
  s += swz_f32<0x101f>(s);
  s += swz_f32<0x081f>(s);
  s += swz_f32<0x041f>(s);
  if (lane == 0) lse[wid] = m + __logf(s);
}

// ---------------------------------------------------------------------------
// Kernel B: CTC forward trellis. One workgroup per batch element.
// 320 threads (10 waves); thread s owns trellis state s (s < 301).
// Logits rows streamed into LDS with TDM double-buffering; own alpha carried
// in a register so each step needs only 2 neighbor LDS reads + 1 write +
// 1 barrier.
// ---------------------------------------------------------------------------
__global__ void __launch_bounds__(320)
W2L_ctc_kernel(const float* __restrict__ logits,
               const int* __restrict__ audio_len,
               const int* __restrict__ labels,
               const float* __restrict__ lse,
               float* __restrict__ loss) {
  const int b = (int)blockIdx.x;
  const int s = (int)threadIdx.x;

  __shared__ float alphaA[S_ + 3];          // state s stored at idx s+2
  __shared__ float alphaB[S_ + 3];
  __shared__ float rowbuf[2][CHUNK * V_];
  __shared__ float lsebuf[CHUNK];
  __shared__ int   sh_cnt;

  // ---- one-time setup -----------------------------------------------------
  if (threadIdx.x == 0) sh_cnt = 0;
  if (threadIdx.x < 2) { alphaA[threadIdx.x] = NEG; alphaB[threadIdx.x] = NEG; }
  __syncthreads();
  if (s < L_ && labels[(size_t)b * L_ + s] >= 0) atomicAdd(&sh_cnt, 1);

  // extended symbol + skip-allowed for this thread's state
  int  e = 0;
  bool allow = false;
  if (s < S_ && (s & 1)) {
    int i  = (s - 1) >> 1;
    int lv = labels[(size_t)b * L_ + i];
    e = lv < 0 ? 0 : lv;
    int ep = -1;
    if (s >= 3) {
      int lv2 = labels[(size_t)b * L_ + i - 1];
      ep = lv2 < 0 ? 0 : lv2;
    }
    allow = (e != 0) && (e != ep);
  }

  // logit_length transform: ceil((a+1)/128) // 2 ; t_final = clip(.,1,T)-1
  const int a_len   = audio_len[b];
  int ll            = ((a_len + 128) >> 7) >> 1;
  ll                = ll < 1 ? 1 : (ll > T_ ? T_ : ll);
  const int total_t = ll;                 // frames used = t_final + 1
  const int nchunks = (total_t + CHUNK - 1) / CHUNK;

  const float* logits_b = logits + (size_t)b * T_ * V_;
  const float* lse_b    = lse + (size_t)b * T_;
  const bool wave0 = (__builtin_amdgcn_readfirstlane((int)threadIdx.x) == 0);

  // prologue: TDM-load chunk 0
  if (wave0) {
    int rows0 = total_t < CHUNK ? total_t : CHUNK;
    tdm_load_rows(logits_b, (unsigned)(__SIZE_TYPE__)&rowbuf[0][0], rows0);
  }

  float* cur = alphaA;
  float* nxt = alphaB;
  float  a_own = NEG;                     // register copy of cur[s+2]

  for (int c = 0; c < nchunks; ++c) {
    const int rows_c = (total_t - c * CHUNK) < CHUNK ? (total_t - c * CHUNK)
                                                     : CHUNK;
    // stage this chunk's lse values into LDS (safe: previous chunk's last
    // step ended with a barrier, so all readers of lsebuf are done)
    if (s < rows_c) lsebuf[s] = lse_b[c * CHUNK + s];

    const bool has_next = (c + 1 < nchunks);
    if (wave0) {
      if (has_next) {
        int rows_n = (total_t - (c + 1) * CHUNK) < CHUNK
                         ? (total_t - (c + 1) * CHUNK) : CHUNK;
        tdm_load_rows(logits_b + (size_t)(c + 1) * CHUNK * V_,
                      (unsigned)(__SIZE_TYPE__)&rowbuf[(c + 1) & 1][0],
                      rows_n);
        __builtin_amdgcn_s_wait_tensorcnt(1);   // chunk c complete
      } else {
        __builtin_amdgcn_s_wait_tensorcnt(0);
      }
    }
    __syncthreads();                      // tile c + lsebuf visible to all

    const float* rb = &rowbuf[c & 1][0];
    for (int r = 0; r < rows_c; ++r) {
      const int t = c * CHUNK + r;
      const float* row = rb + r * V_;
      const float lse_t = lsebuf[r];
      if (t == 0) {
        if (s < S_) {
          float lp = row[e] - lse_t;
          a_own = (s < 2) ? lp : NEG;
          cur[s + 2] = a_own;
        }
        __syncthreads();
      } else {
        if (s < S_) {
          float lp = row[e] - lse_t;
          float a0 = a_own;               // own alpha from register
          float a1 = cur[s + 1];
          float a2 = allow ? cur[s] : NEG;
          float m  = fmaxf(a0, fmaxf(a1, a2));
          float sm = __expf(a0 - m) + __expf(a1 - m) + __expf(a2 - m);
          a_own = lp + m + __logf(sm);
          nxt[s + 2] = a_own;
        }
        __syncthreads();                  // single barrier per step
        float* tmp = cur; cur = nxt; nxt = tmp;
      }
    }
  }

  // readout at final frame: logaddexp(alpha[2*len], alpha[2*len-1])
  if (threadIdx.x == 0) {
    int s_last = 2 * sh_cnt;
    float ab = cur[s_last + 2];
    int sl1  = s_last - 1; if (sl1 < 0) sl1 = 0;
    float al = cur[sl1 + 2];
    float m  = fmaxf(ab, al);
    loss[b] = -(m + __logf(__expf(ab - m) + __expf(al - m)));
  }
}

// ---------------------------------------------------------------------------
// Kernel C: mean over batch -> scalar output
// ---------------------------------------------------------------------------
__global__ void W2L_mean_kernel(const float* __restrict__ loss,
                                float* __restrict__ out) {
  if (threadIdx.x == 0) {
    float acc = 0.f;
    for (int i = 0; i < B_; ++i) acc += loss[i];
    out[0] = acc / (float)B_;
  }
}

// ---------------------------------------------------------------------------
extern "C" void kernel_launch(void* const* d_in, const int* in_sizes, int n_in,
                              void* d_out, int out_size, void* d_ws,
                              size_t ws_size, hipStream_t stream) {
  const float* logits = (const float*)d_in[0];
  const int*   audio  = (const int*)d_in[1];
  const int*   labels = (const int*)d_in[2];

  float* lse  = (float*)d_ws;                 // B*T floats
  float* loss = lse + (size_t)B_ * T_;        // B floats
  float* out  = (float*)d_out;

  // A: 64000 rows, 8 waves/block -> 8000 blocks
  W2L_lse_kernel<<<(B_ * T_) / 8, 256, 0, stream>>>(logits, lse);
  // B: one workgroup per batch element
  W2L_ctc_kernel<<<B_, 320, 0, stream>>>(logits, audio, labels, lse, loss);
  // C: mean
  W2L_mean_kernel<<<1, 32, 0, stream>>>(loss, out);
}